// SMSELossFunc_40819369181889
// MI455X (gfx1250) — compile-verified
//
#include <hip/hip_runtime.h>

// Problem geometry (L=16, C=3, H=512, W=512), f32 everywhere.
#define NCH     48        // L*C independent channels
#define HW      262144    // H*W elements per channel
#define HW4     65536     // HW / 4 (float4 groups per channel)
#define CHUNKS  32        // blocks per channel
#define TPB     256       // threads per block (8 wave32)
#define ITERS   ((HW4 / CHUNKS) / TPB)   // = 8 float4 loads per thread per tensor
#define V_EPS   1e-5f

typedef float v2f __attribute__((ext_vector_type(2)));
typedef float v4f __attribute__((ext_vector_type(4)));
typedef float v8f __attribute__((ext_vector_type(8)));

// ---------------------------------------------------------------------------
// Stage 1: streaming pass. Each block owns 1/32 of one channel, accumulates
// (S_ee, S_ce, S_cc, S_m) per lane with 128-bit NT loads, folds across the
// wave with V_WMMA_F32_16X16X4_F32, then across waves in LDS, and writes a
// 4-float partial per block to the workspace. Fully deterministic.
// ---------------------------------------------------------------------------
__global__ __launch_bounds__(TPB) void smse_stage1(
    const float* __restrict__ corr,
    const float* __restrict__ est,
    const float* __restrict__ mask,
    float* __restrict__ partial)
{
    const int blk   = blockIdx.x;
    const int ch    = blk / CHUNKS;
    const int chunk = blk % CHUNKS;

    const v4f* __restrict__ c4 = (const v4f*)corr + (size_t)ch * HW4;
    const v4f* __restrict__ e4 = (const v4f*)est  + (size_t)ch * HW4;
    const v4f* __restrict__ m4 = (const v4f*)mask + (size_t)ch * HW4;

    int j = chunk * (HW4 / CHUNKS) + threadIdx.x;

    float ee = 0.f, ce = 0.f, cc = 0.f, sm = 0.f;
#pragma unroll
    for (int i = 0; i < ITERS; ++i, j += TPB) {
        v4f c = __builtin_nontemporal_load(c4 + j);   // zero reuse -> NT hint
        v4f e = __builtin_nontemporal_load(e4 + j);
        v4f m = __builtin_nontemporal_load(m4 + j);
#pragma unroll
        for (int k = 0; k < 4; ++k) {
            float em = e[k] * m[k];
            ee = fmaf(e[k], em, ee);          // S_ee += e^2 m
            ce = fmaf(c[k], em, ce);          // S_ce += c e m
            cc = fmaf(c[k], c[k] * m[k], cc); // S_cc += c^2 m
            sm += m[k];                       // S_m  += m
        }
    }

    // ---- wave32 reduction via WMMA f32 16x16x4 ----
    // A layout (16x4 f32): lanes 0-15 supply A[l][0..1], lanes 16-31 A[l-16][2..3].
    // B selection: B[0][0]=B[2][0]=1, B[1][1]=B[3][1]=1, else 0
    //   => D[m][0] = A[m][0]+A[m][2] (stat.x folded over lane,lane+16)
    //      D[m][1] = A[m][1]+A[m][3] (stat.y folded)
    const int lane = threadIdx.x & 31;
    const int wave = threadIdx.x >> 5;

    v2f a1; a1.x = ee; a1.y = ce;   // WMMA #1 folds (ee, ce)
    v2f a2; a2.x = cc; a2.y = sm;   // WMMA #2 folds (cc, sm)
    v2f b;
    b.x = (lane == 0 || lane == 16) ? 1.f : 0.f;   // B[0][0], B[2][0]
    b.y = (lane == 1 || lane == 17) ? 1.f : 0.f;   // B[1][1], B[3][1]
    v8f z = {};

    v8f d1 = __builtin_amdgcn_wmma_f32_16x16x4_f32(false, a1, false, b,
                                                   (short)0, z, false, false);
    v8f d2 = __builtin_amdgcn_wmma_f32_16x16x4_f32(false, a2, false, b,
                                                   (short)0, z, false, false);

    // D VGPR v holds row M=v (lanes 0-15, N=lane) and M=v+8 (lanes 16-31).
    // Horizontal-add the 8 rows held by this lane, then fold lane^16:
    float h1 = d1[0] + d1[1] + d1[2] + d1[3] + d1[4] + d1[5] + d1[6] + d1[7];
    float h2 = d2[0] + d2[1] + d2[2] + d2[3] + d2[4] + d2[5] + d2[6] + d2[7];
    h1 += __shfl_xor(h1, 16, 32);
    h2 += __shfl_xor(h2, 16, 32);
    // Now: lane 0 -> h1 = total ee, h2 = total cc
    //      lane 1 -> h1 = total ce, h2 = total sm

    __shared__ float lds[TPB / 32][4];
    if (lane == 0) { lds[wave][0] = h1; lds[wave][2] = h2; }
    if (lane == 1) { lds[wave][1] = h1; lds[wave][3] = h2; }
    __syncthreads();

    if (threadIdx.x < 4) {
        float s = 0.f;
#pragma unroll
        for (int w = 0; w < TPB / 32; ++w) s += lds[w][threadIdx.x];
        partial[blk * 4 + threadIdx.x] = s;
    }
}

// ---------------------------------------------------------------------------
// Stage 2: single tiny block. Combine 32 chunk-partials per (channel, stat),
// apply alpha / eps logic, err = Scc - 2a*Sce + a^2*See, ssq per l, mean /3.
// ---------------------------------------------------------------------------
__global__ __launch_bounds__(TPB) void smse_stage2(
    const float* __restrict__ partial, float* __restrict__ out)
{
    __shared__ float stat[NCH][4];
    __shared__ float ssq[16];
    const int t = threadIdx.x;

    if (t < NCH * 4) {
        const int ch = t >> 2, s = t & 3;
        float sum = 0.f;
        for (int k = 0; k < CHUNKS; ++k)
            sum += partial[(ch * CHUNKS + k) * 4 + s];
        stat[ch][s] = sum;
    }
    __syncthreads();

    if (t < 16) {
        float acc = 0.f;
        for (int c = 0; c < 3; ++c) {
            const int ch = t * 3 + c;
            const float ee = stat[ch][0], ce = stat[ch][1];
            const float cc = stat[ch][2], sm = stat[ch][3];
            const float alpha = (ee > V_EPS) ? (ce / ee) : 0.f;
            const float err = cc - 2.f * alpha * ce + alpha * alpha * ee;
            acc += err / sm;
        }
        ssq[t] = acc;
    }
    __syncthreads();

    if (t == 0) {
        float r = 0.f;
        for (int l = 0; l < 16; ++l) r += ssq[l];
        out[0] = r * (1.0f / 16.0f) * (1.0f / 3.0f);
    }
}

// ---------------------------------------------------------------------------
extern "C" void kernel_launch(void* const* d_in, const int* in_sizes, int n_in,
                              void* d_out, int out_size, void* d_ws, size_t ws_size,
                              hipStream_t stream) {
    const float* corr = (const float*)d_in[0];
    const float* est  = (const float*)d_in[1];
    const float* mask = (const float*)d_in[2];
    float* partial    = (float*)d_ws;      // needs NCH*CHUNKS*4 floats = 24 KB

    smse_stage1<<<NCH * CHUNKS, TPB, 0, stream>>>(corr, est, mask, partial);
    smse_stage2<<<1, TPB, 0, stream>>>(partial, (float*)d_out);
}